// S6_56581899157850
// MI455X (gfx1250) — compile-verified
//
#include <hip/hip_runtime.h>
#include <math.h>

// Problem constants (match reference)
#define BB   4
#define LSEQ 1024
#define DD   512
#define NN   16
#define MM   (BB * LSEQ)   // 4096 rows
#define CHUNK 256          // timesteps of Bm/Cm staged in LDS per pass

typedef float v2f __attribute__((ext_vector_type(2)));
typedef float v8f __attribute__((ext_vector_type(8)));

#define ACT_NONE 0
#define ACT_SOFTPLUS 1

// ---------------------------------------------------------------------------
// C[M,Nn] = act( A[M,K] @ W[K,Nn] + bias[Nn] )   via V_WMMA_F32_16X16X4_F32.
//
// Wave macro-tile: 32(M) x (16*NSUB)(N); acc = 2 x NSUB v8f fragments.
// Compile-time K/Nn -> immediate-offset loads + pointer bumps (no per-iter
// 64-bit address rebuilds).  Per k-step: 2 b64 A-loads + 2*NSUB b32 B-loads
// feed 2*NSUB WMMAs (1.25 loads/WMMA at NSUB=4).
//
// Fragment layouts (ISA 7.12.2, f32):
//   A 16x4 : lane m = lane&15; VGPR0 = K=(lane>=16?2:0), VGPR1 = K+1
//   B 4x16 : lane n = lane&15; same K striping across the 2 VGPRs
//   D 16x16: VGPR r -> row r + 8*(lane>=16), col = lane&15
// ---------------------------------------------------------------------------
template <int KK, int NnT, int NSUB, int ACT>
__global__ __launch_bounds__(128) void gemm_f32_wmma(
    const float* __restrict__ A, const float* __restrict__ W,
    const float* __restrict__ bias, float* __restrict__ C)
{
  const int lane = threadIdx.x & 31;
  const int wave = threadIdx.x >> 5;
  const int m0 = (blockIdx.x * 4 + wave) * 32;       // 32 rows per wave
  const int n0 = blockIdx.y * (16 * NSUB);

  const int mrow = lane & 15;
  const int ncol = lane & 15;
  const int koff = (lane >> 4) << 1;                 // 0 | 2

  const float* __restrict__ ap0 = A + (size_t)(m0 + mrow) * KK + koff;
  const float* __restrict__ ap1 = ap0 + (size_t)16 * KK;
  const float* __restrict__ bp  = W + (size_t)koff * NnT + n0 + ncol;

  v8f acc[2][NSUB];
  #pragma unroll
  for (int t = 0; t < 2; ++t)
    #pragma unroll
    for (int j = 0; j < NSUB; ++j) acc[t][j] = (v8f){};

  for (int k0 = 0; k0 < KK; k0 += 4) {
    const v2f a0 = *(const v2f*)ap0;                 // A[m,      k+koff..+1]
    const v2f a1 = *(const v2f*)ap1;                 // A[m+16,   k+koff..+1]
    v2f bf[NSUB];
    #pragma unroll
    for (int j = 0; j < NSUB; ++j) {
      bf[j].x = bp[j * 16];                          // W[k+koff,   n + 16j]
      bf[j].y = bp[j * 16 + NnT];                    // W[k+koff+1, n + 16j]
    }
    #pragma unroll
    for (int j = 0; j < NSUB; ++j) {
      acc[0][j] = __builtin_amdgcn_wmma_f32_16x16x4_f32(
          false, a0, false, bf[j], (short)0, acc[0][j], false, false);
      acc[1][j] = __builtin_amdgcn_wmma_f32_16x16x4_f32(
          false, a1, false, bf[j], (short)0, acc[1][j], false, false);
    }
    ap0 += 4; ap1 += 4; bp += (size_t)4 * NnT;
  }

  #pragma unroll
  for (int j = 0; j < NSUB; ++j) {
    const int nc = n0 + j * 16 + ncol;
    const float bv = bias[nc];
    #pragma unroll
    for (int t = 0; t < 2; ++t) {
      #pragma unroll
      for (int r = 0; r < 8; ++r) {
        const int mr = m0 + t * 16 + ((lane < 16) ? r : (r + 8));
        float v = acc[t][j][r] + bv;
        if (ACT == ACT_SOFTPLUS)                      // stable softplus
          v = fmaxf(v, 0.0f) + log1pf(expf(-fabsf(v)));
        C[(size_t)mr * NnT + nc] = v;
      }
    }
  }
}

// ---------------------------------------------------------------------------
// Fused selective-scan + output contraction + skip + ReLU.
// One thread owns one (b,d) channel: 16 SSM states in registers, sequential
// over L.  Bm/Cm (shared across d for a given (b,l)) staged in LDS in
// CHUNK-sized tiles.  (B,L,D,N) tensors never touch HBM (saves ~1.6 GB).
// yr[b,l,d] = relu( sum_n h[n]*Cm[l,n] + Dskip[d]*y )
//   with h[n] = exp(dt*A[d,n]) * h[n] + (dt*y)*Bm[l,n]
// ---------------------------------------------------------------------------
__global__ __launch_bounds__(256) void s6_scan_kernel(
    const float* __restrict__ y,  const float* __restrict__ dt,
    const float* __restrict__ Bm, const float* __restrict__ Cm,
    const float* __restrict__ Aln, const float* __restrict__ Dskip,
    float* __restrict__ yr)
{
  __shared__ float sB[CHUNK * NN];
  __shared__ float sC[CHUNK * NN];

  const int d = blockIdx.x * 256 + threadIdx.x;
  const int b = blockIdx.y;

  float Ad[NN];
  #pragma unroll
  for (int n = 0; n < NN; ++n) Ad[n] = Aln[d * NN + n];
  const float dsk = Dskip[d];

  float h[NN];
  #pragma unroll
  for (int n = 0; n < NN; ++n) h[n] = 0.0f;

  for (int l0 = 0; l0 < LSEQ; l0 += CHUNK) {
    const float* __restrict__ Bsrc = Bm + ((size_t)b * LSEQ + l0) * NN;
    const float* __restrict__ Csrc = Cm + ((size_t)b * LSEQ + l0) * NN;
    for (int t = threadIdx.x; t < CHUNK * NN; t += 256) {
      sB[t] = Bsrc[t];
      sC[t] = Csrc[t];
    }
    __syncthreads();

    for (int lc = 0; lc < CHUNK; ++lc) {
      const size_t idx = ((size_t)b * LSEQ + (l0 + lc)) * DD + d;
      const float dtv = dt[idx];
      const float yv  = y[idx];
      const float dty = dtv * yv;
      float accv = 0.0f;
      #pragma unroll
      for (int n = 0; n < NN; ++n) {
        const float a = expf(dtv * Ad[n]);            // ZOH discretization
        h[n] = fmaf(a, h[n], dty * sB[lc * NN + n]);
        accv = fmaf(h[n], sC[lc * NN + n], accv);
      }
      const float outv = fmaf(dsk, yv, accv);
      yr[idx] = fmaxf(outv, 0.0f);                    // ReLU
    }
    __syncthreads();
  }
}

// ---------------------------------------------------------------------------
extern "C" void kernel_launch(void* const* d_in, const int* in_sizes, int n_in,
                              void* d_out, int out_size, void* d_ws, size_t ws_size,
                              hipStream_t stream) {
  (void)in_sizes; (void)n_in; (void)out_size; (void)ws_size;

  const float* x    = (const float*)d_in[0];   // (B,L,D)
  const float* A    = (const float*)d_in[1];   // (NL,D,N)
  const float* Dsk  = (const float*)d_in[2];   // (NL,D)
  const float* WB   = (const float*)d_in[3];   // (NL,D,N)
  const float* bB   = (const float*)d_in[4];   // (NL,N)
  const float* WC   = (const float*)d_in[5];   // (NL,D,N)
  const float* bC   = (const float*)d_in[6];   // (NL,N)
  const float* Wdt  = (const float*)d_in[7];   // (NL,D,D)
  const float* bdt  = (const float*)d_in[8];   // (NL,D)
  const float* Wlin = (const float*)d_in[9];   // (NL,D,D)
  const float* blin = (const float*)d_in[10];  // (NL,D)
  const float* Wdec = (const float*)d_in[11];  // (D,OUT)
  const float* bdec = (const float*)d_in[12];  // (OUT,)
  float* out = (float*)d_out;                  // (B,L,OUT)

  // Workspace carve-up (floats): 3 big (M*D) + Bm/Cm (M*N)  ~= 25.7 MB
  float* ws = (float*)d_ws;
  const size_t big = (size_t)MM * DD;
  float* yr  = ws;                 // scan output (post-ReLU)
  float* yb  = ws + big;           // layer activations
  float* dtb = ws + 2 * big;       // softplus(dt) buffer
  float* Bmb = ws + 3 * big;       // (M,N)
  float* Cmb = Bmb + (size_t)MM * NN;

  const dim3 blk(128);
  // Big GEMM (Nn=512): 32x64 per wave -> grid (MM/128, 512/64)
  const dim3 gBig(MM / 128, DD / 64);
  // Small GEMM (Nn=16): 32x16 per wave -> grid (MM/128, 1)
  const dim3 gSmall(MM / 128, 1);
  const dim3 gScan(DD / 256, BB);

  const float* cur = x;
  for (int i = 0; i < 2; ++i) {
    // dt = softplus(cur @ Wdt[i] + bdt[i])
    gemm_f32_wmma<DD, DD, 4, ACT_SOFTPLUS><<<gBig, blk, 0, stream>>>(
        cur, Wdt + (size_t)i * DD * DD, bdt + (size_t)i * DD, dtb);
    // Bm = cur @ WB[i] + bB[i] ; Cm = cur @ WC[i] + bC[i]
    gemm_f32_wmma<DD, NN, 1, ACT_NONE><<<gSmall, blk, 0, stream>>>(
        cur, WB + (size_t)i * DD * NN, bB + (size_t)i * NN, Bmb);
    gemm_f32_wmma<DD, NN, 1, ACT_NONE><<<gSmall, blk, 0, stream>>>(
        cur, WC + (size_t)i * DD * NN, bC + (size_t)i * NN, Cmb);
    // fused recurrence + C-contraction + skip + ReLU
    s6_scan_kernel<<<gScan, 256, 0, stream>>>(
        cur, dtb, Bmb, Cmb, A + (size_t)i * DD * NN, Dsk + (size_t)i * DD, yr);
    // y = yr @ Wlin[i] + blin[i]
    gemm_f32_wmma<DD, DD, 4, ACT_NONE><<<gBig, blk, 0, stream>>>(
        yr, Wlin + (size_t)i * DD * DD, blin + (size_t)i * DD, yb);
    cur = yb;
  }
  // decoder: out = cur @ Wdec + bdec
  gemm_f32_wmma<DD, DD, 4, ACT_NONE><<<gBig, blk, 0, stream>>>(cur, Wdec, bdec, out);
}